// SVDTokenObserver_21680994910749
// MI455X (gfx1250) — compile-verified
//
#include <hip/hip_runtime.h>
#include <hip/hip_bf16.h>

typedef float v2f __attribute__((ext_vector_type(2)));
typedef float v8f __attribute__((ext_vector_type(8)));

#define SEQ 5
#define DIM 512

// ---------------------------------------------------------------------------
// Kernel A: batched 5x5 Gram matrices via V_WMMA_F32_16X16X4_F32.
// One wave32 handles 3 batches packed into a 16x16 tile (rows 0..14 used).
// G[b] is written into the vh slot of d_out (overwritten later by kernel B).
// ---------------------------------------------------------------------------
__global__ void gram_wmma_kernel(const float* __restrict__ x,
                                 float* __restrict__ gout, int nbatch)
{
    const int tid  = threadIdx.x;
    const int lane = tid & 31;
    const long long wave = (long long)blockIdx.x * (blockDim.x >> 5) + (tid >> 5);
    const long long b0   = wave * 3;

    // A-matrix layout (32-bit, 16x4): lane m=lane&15 holds row M=m;
    // lanes 0-15: K slots {0,1}; lanes 16-31: K slots {2,3} (VGPR0,VGPR1).
    const int m    = lane & 15;
    const int half = lane >> 4;
    const int g    = m / 5;          // batch within wave (3 valid, m==15 -> 3)
    const int s    = m - 5 * g;

    const bool rowValid = (m < 15) && (b0 + g < (long long)nbatch);
    const long long batch = rowValid ? (b0 + g) : 0;  // invalid lanes read x[0]: safe,
                                                      // garbage lands in ignored C entries
    const float* row = x + (batch * SEQ + (rowValid ? s : 0)) * (long long)DIM
                         + 2 * half;

    v8f c = {};
    #pragma unroll 4
    for (int k = 0; k < DIM; k += 4) {
        v2f a = *(const v2f*)(row + k);   // 8B aligned (k%4==0, +2*half)
        // Gram: B = A^T -> identical register image; pass same regs for A and B.
        c = __builtin_amdgcn_wmma_f32_16x16x4_f32(
                false, a, false, a, (short)0, c, false, false);
    }

    // C/D layout: VGPR j holds row M = j + 8*half, column N = lane&15.
    float carr[8];
    *(v8f*)carr = c;
    const int N = lane & 15;
    if (N < 15) {
        const int gB = N / 5;
        const int t  = N - 5 * gB;
        const long long bb = b0 + gB;
        if (bb < (long long)nbatch) {
            #pragma unroll
            for (int j = 0; j < 8; ++j) {
                const int M    = j + 8 * half;
                const int srow = M - 5 * gB;
                if (srow >= 0 && srow < 5)
                    gout[bb * 25 + srow * 5 + t] = carr[j];
            }
        }
    }
}

// ---------------------------------------------------------------------------
// Kernel B: per-batch 5x5 symmetric eigensolve (cyclic Jacobi, registers only),
// descending sort, then sv / vh / features exactly as the reference.
// ---------------------------------------------------------------------------
__device__ __forceinline__ float fin_or(float x, float alt) {
    return __builtin_isfinite(x) ? x : alt;
}

__global__ void jacobi_kernel(float* __restrict__ out, int nbatch)
{
    const int b = blockIdx.x * blockDim.x + threadIdx.x;
    if (b >= nbatch) return;

    float* gptr = out + (long long)nbatch * SEQ + (long long)b * 25;

    float a[5][5], v[5][5];
    #pragma unroll
    for (int i = 0; i < 5; ++i)
        #pragma unroll
        for (int j = 0; j < 5; ++j) {
            a[i][j] = gptr[i * 5 + j];
            v[i][j] = (i == j) ? 1.0f : 0.0f;
        }

    // Cyclic Jacobi sweeps; p,q,k fully unrolled => arrays stay in VGPRs.
    for (int sweep = 0; sweep < 8; ++sweep) {
        #pragma unroll
        for (int p = 0; p < 4; ++p) {
            #pragma unroll
            for (int q = p + 1; q < 5; ++q) {
                const float apq = a[p][q];
                if (fabsf(apq) > 1e-30f) {
                    const float theta = (a[q][q] - a[p][p]) / (2.0f * apq);
                    float t = 1.0f / (fabsf(theta) + sqrtf(theta * theta + 1.0f));
                    t = (theta < 0.0f) ? -t : t;
                    const float cc = rsqrtf(t * t + 1.0f);
                    const float ss = t * cc;
                    #pragma unroll
                    for (int k = 0; k < 5; ++k) {       // A := A * R
                        const float t1 = a[k][p], t2 = a[k][q];
                        a[k][p] = cc * t1 - ss * t2;
                        a[k][q] = ss * t1 + cc * t2;
                    }
                    #pragma unroll
                    for (int k = 0; k < 5; ++k) {       // A := R^T * A
                        const float t1 = a[p][k], t2 = a[q][k];
                        a[p][k] = cc * t1 - ss * t2;
                        a[q][k] = ss * t1 + cc * t2;
                    }
                    #pragma unroll
                    for (int k = 0; k < 5; ++k) {       // V := V * R
                        const float t1 = v[k][p], t2 = v[k][q];
                        v[k][p] = cc * t1 - ss * t2;
                        v[k][q] = ss * t1 + cc * t2;
                    }
                }
            }
        }
    }

    float w[5];
    #pragma unroll
    for (int i = 0; i < 5; ++i) w[i] = a[i][i];

    // selection sort, descending eigenvalues; swap V columns along
    #pragma unroll
    for (int i = 0; i < 4; ++i)
        #pragma unroll
        for (int j = i + 1; j < 5; ++j)
            if (w[j] > w[i]) {
                const float tw = w[i]; w[i] = w[j]; w[j] = tw;
                #pragma unroll
                for (int k = 0; k < 5; ++k) {
                    const float tv = v[k][i]; v[k][i] = v[k][j]; v[k][j] = tv;
                }
            }

    // sv = clip(sqrt(clip(w,0)), 1e-6), finite-guarded
    float sv[5];
    #pragma unroll
    for (int i = 0; i < 5; ++i) {
        float sval = sqrtf(fmaxf(w[i], 0.0f));
        sval = fmaxf(sval, 1e-6f);
        sv[i] = fin_or(sval, 1.0f);
    }

    // vh = V^T, finite-guarded with 0
    float vh[5][5];
    #pragma unroll
    for (int i = 0; i < 5; ++i)
        #pragma unroll
        for (int j = 0; j < 5; ++j)
            vh[i][j] = fin_or(v[j][i], 0.0f);

    float svsum = 0.0f;
    #pragma unroll
    for (int i = 0; i < 5; ++i) svsum += sv[i];
    const float inv = 1.0f / (svsum + 1e-8f);

    float s_norm[5];
    #pragma unroll
    for (int i = 0; i < 5; ++i) s_norm[i] = sv[i] * inv;

    float sumsq = 0.0f, diagsq = 0.0f, ent = 0.0f;
    #pragma unroll
    for (int i = 0; i < 5; ++i) {
        #pragma unroll
        for (int j = 0; j < 5; ++j) sumsq += vh[i][j] * vh[i][j];
        diagsq += vh[i][i] * vh[i][i];
        ent    -= s_norm[i] * logf(fmaxf(s_norm[i], 1e-8f));
    }
    const float offd = fmaxf(sumsq - diagsq, 0.0f);

    // ---- stores (sv | vh | features), features finite-guarded ----
    float* svp = out + (long long)b * 5;
    #pragma unroll
    for (int i = 0; i < 5; ++i) svp[i] = sv[i];

    #pragma unroll
    for (int i = 0; i < 5; ++i)
        #pragma unroll
        for (int j = 0; j < 5; ++j)
            gptr[i * 5 + j] = vh[i][j];     // overwrites the stashed Gram

    float* fp = out + (long long)nbatch * 30 + (long long)b * 12;
    #pragma unroll
    for (int i = 0; i < 5; ++i) fp[i]     = fin_or(s_norm[i], 0.0f);
    #pragma unroll
    for (int i = 0; i < 5; ++i) fp[5 + i] = fin_or(vh[i][i], 0.0f);
    fp[10] = fin_or(offd, 0.0f);
    fp[11] = fin_or(ent, 0.0f);
}

// ---------------------------------------------------------------------------
extern "C" void kernel_launch(void* const* d_in, const int* in_sizes, int n_in,
                              void* d_out, int out_size, void* d_ws, size_t ws_size,
                              hipStream_t stream) {
    (void)n_in; (void)out_size; (void)d_ws; (void)ws_size;
    const float* x = (const float*)d_in[0];
    float* out = (float*)d_out;
    const int nb = in_sizes[0] / (SEQ * DIM);   // 16384

    const int threadsA = 256;                   // 8 waves/block, 3 batches/wave
    const int wavesNeeded = (nb + 2) / 3;
    const int blocksA = (wavesNeeded + (threadsA / 32) - 1) / (threadsA / 32);
    gram_wmma_kernel<<<blocksA, threadsA, 0, stream>>>(
        x, out + (long long)nb * SEQ, nb);

    const int threadsB = 256;
    const int blocksB = (nb + threadsB - 1) / threadsB;
    jacobi_kernel<<<blocksB, threadsB, 0, stream>>>(out, nb);
}